// ForwardForwardNodeEdgeCoutingAutoencoder_19593640804424
// MI455X (gfx1250) — compile-verified
//
#include <hip/hip_runtime.h>
#include <hip/hip_bf16.h>
#include <stdint.h>

// Fuzzy min/max layer:
//   out[n,j] = reduce_{i : edges[n,j,i]==1} x[n,i]   (min if op_idx[j]==0 else max)
//   if no edge selected: out[n,j] = x[n, fix_idx[n,j]]
// Memory-bound: edges streams dominate (512MB per layer, ~1.07GB total -> ~46us
// floor at 23.3 TB/s). NT b128 loads for edges, CDNA5 async-to-LDS staging for
// the x row, wave32-cooperative shfl reduction. Unroll 4 on the output loop to
// keep >=4 independent b128 NT loads in flight per wave.

#define SENTINEL -3.402823e38f
#define NOEDGE_THRESH -1.0e30f

typedef int   v4i __attribute__((ext_vector_type(4)));
typedef float v4f __attribute__((ext_vector_type(4)));

template <int F_IN_T, int F_OUT_T>
__global__ __launch_bounds__(256) void fuzzy_layer_kernel(
    const float* __restrict__ x,      // [N, F_IN_T]
    const int*   __restrict__ op_idx, // [F_OUT_T]
    const int*   __restrict__ edges,  // [N, F_OUT_T, F_IN_T]
    const int*   __restrict__ fixidx, // [N, F_OUT_T]
    float*       __restrict__ out)    // [N, F_OUT_T]
{
    constexpr int WAVES = 8;                 // 256 threads, wave32
    constexpr int PER_LANE = F_IN_T / 128;   // v4i chunks per lane (256->2, 128->1)

    const int n    = blockIdx.x;
    const int tid  = threadIdx.x;
    const int wave = tid >> 5;
    const int lane = tid & 31;

    __shared__ float xs[F_IN_T];

    // ---- Stage x[n, :] into LDS via CDNA5 async load-to-LDS (gfx1250) ----
    const float* xrow = x + (size_t)n * F_IN_T;
    if (tid < F_IN_T) {
        uint32_t lds_off = (uint32_t)(uintptr_t)(&xs[tid]);           // LDS aperture: low 32b = LDS offset
        unsigned long long gaddr = (unsigned long long)(xrow + tid);
        asm volatile("global_load_async_to_lds_b32 %0, %1, off"
                     :
                     : "v"(lds_off), "v"(gaddr)
                     : "memory");
    }
#if __has_builtin(__builtin_amdgcn_s_wait_asynccnt)
    __builtin_amdgcn_s_wait_asynccnt(0);
#else
    asm volatile("s_wait_asynccnt 0" ::: "memory");
#endif
    __syncthreads();

    // ---- One wave per output j; lanes split the F_IN_T reduction ----
    const size_t row_base = (size_t)n * F_OUT_T;

#pragma unroll 4
    for (int j = wave; j < F_OUT_T; j += WAVES) {
        // op is uniform across the wave -> scalar load
        const bool  is_min = (op_idx[j] == 0);
        const float s      = is_min ? -1.0f : 1.0f;   // min(a,b) = -max(-a,-b)

        const v4i* ep = (const v4i*)(edges + (row_base + j) * F_IN_T) + lane;

        float v = SENTINEL;
#pragma unroll
        for (int c = 0; c < PER_LANE; ++c) {
            v4i e = __builtin_nontemporal_load(ep + c * 32);   // coalesced 512B/wave, NT: don't pollute L2
            const int base = (c * 32 + lane) * 4;
            v4f xv = *(const v4f*)&xs[base];
            v = fmaxf(v, e.x ? s * xv.x : SENTINEL);
            v = fmaxf(v, e.y ? s * xv.y : SENTINEL);
            v = fmaxf(v, e.z ? s * xv.z : SENTINEL);
            v = fmaxf(v, e.w ? s * xv.w : SENTINEL);
        }

        // wave32 max-reduction
        for (int m = 16; m > 0; m >>= 1)
            v = fmaxf(v, __shfl_xor(v, m, 32));

        if (lane == 0) {
            float r;
            if (v <= NOEDGE_THRESH) {
                // no sampled edge: forced edge at fix_idx -> result is that x value
                r = xs[fixidx[row_base + j]];
            } else {
                r = s * v;
            }
            out[row_base + j] = r;
        }
    }
}

extern "C" void kernel_launch(void* const* d_in, const int* in_sizes, int n_in,
                              void* d_out, int out_size, void* d_ws, size_t ws_size,
                              hipStream_t stream) {
    constexpr int F_IN = 256, F_HID = 128;

    const float* x        = (const float*)d_in[0];   // [N, 256]
    const int*   op_idx1  = (const int*)d_in[1];     // [128]
    const int*   op_idx2  = (const int*)d_in[2];     // [256]
    const int*   edges1   = (const int*)d_in[3];     // [N, 128, 256]
    const int*   edges2   = (const int*)d_in[4];     // [N, 256, 128]
    const int*   fix_idx1 = (const int*)d_in[5];     // [N, 128]
    const int*   fix_idx2 = (const int*)d_in[6];     // [N, 256]

    float* out = (float*)d_out;                      // [N, 256]
    float* h   = (float*)d_ws;                       // [N, 128] intermediate (2 MB)

    const int N = in_sizes[0] / F_IN;                // 4096

    // Layer 1: [N,256] -> [N,128]
    fuzzy_layer_kernel<F_IN, F_HID><<<N, 256, 0, stream>>>(x, op_idx1, edges1, fix_idx1, h);
    // Layer 2: [N,128] -> [N,256]
    fuzzy_layer_kernel<F_HID, F_IN><<<N, 256, 0, stream>>>(h, op_idx2, edges2, fix_idx2, out);
}